// PortHamiltonianNeuralNetwork_11235634446928
// MI455X (gfx1250) — compile-verified
//
#include <hip/hip_runtime.h>

#define NTRAJ   2048
#define T_STEPS 256
#define NS      16
#define HID     128
#define HF      64
#define NPOS    8

typedef __attribute__((ext_vector_type(2))) float v2f;
typedef __attribute__((ext_vector_type(8))) float v8f;

__device__ __forceinline__ v8f wmma_f32_k4(v2f a, v2f b, v8f c) {
  // D = A(16x4 f32) * B(4x16 f32) + C(16x16 f32), full f32 precision
  return __builtin_amdgcn_wmma_f32_16x16x4_f32(false, a, false, b, (short)0, c, false, false);
}

// Branch-free tanh on the transcendental pipe:
// tanh(x) = 1 - 2/(exp(2x)+1) = 1 - 2*rcp(exp2(2*log2e*x)+1)
// Saturates correctly: exp2->inf => +1, exp2->0 => -1. No EXEC divergence.
__device__ __forceinline__ float fast_tanh(float x) {
  float e = __builtin_amdgcn_exp2f(x * 2.8853900817779268f);
  float r = __builtin_amdgcn_rcpf(e + 1.0f);
  return fmaf(-2.0f, r, 1.0f);
}

__global__ __launch_bounds__(256)
void phnn_kernel(const float* __restrict__ u0s,
                 const float* __restrict__ ts,
                 const float* __restrict__ W1,
                 const float* __restrict__ b1,
                 const float* __restrict__ W2,
                 const float* __restrict__ b2,
                 const float* __restrict__ W3,
                 const float* __restrict__ Wf1,
                 const float* __restrict__ bf1,
                 const float* __restrict__ Wf2,
                 const float* __restrict__ bf2,
                 const float* __restrict__ Nv,
                 float* __restrict__ out)
{
  // Padded LDS buffers (pads chosen for conflict-free / 8B-aligned pair loads)
  __shared__ float sW1[NS * 132];                    // W1 row-major, row stride 132
  __shared__ float sb1[HID], sb2[HID], sw3[HID];
  __shared__ float sN[NPOS];
  __shared__ float sTs[T_STEPS];                     // t_sample cache
  __shared__ float sFall[(T_STEPS - 1) * 2 * NPOS];  // forcing table: [step][which][j]
  __shared__ float sUU[2][NS * 18];                  // [0]=u, [1]=u_mid, row stride 18
  __shared__ float sH1[NS * 132];                    // h1, later overwritten with d1
  __shared__ float sD2[NS * 132];                    // d2
  __shared__ float sP[8][NS * 17];                   // per-wave dH partials
  __shared__ float sDH[NS * 17];                     // reduced dH

  const int tid  = threadIdx.x;
  const int wave = tid >> 5;               // this wave's N-tile (0..7)
  const int lane = tid & 31;
  const int m16  = lane & 15;
  const int h2x  = (lane >> 4) << 1;       // k offset within 4-chunk: 0 or 2
  const int row8 = (lane >> 4) << 3;       // C/D row base: 0 or 8
  const int colW = wave * 16 + m16;        // hidden column owned by this lane
  const int trajBase = blockIdx.x * 16;

  // ---- preload constants into LDS ----
  for (int i = tid; i < NS * HID; i += 256) sW1[(i >> 7) * 132 + (i & 127)] = W1[i];
  for (int i = tid; i < HID; i += 256) { sb1[i] = b1[i]; sb2[i] = b2[i]; sw3[i] = W3[i]; }
  if (tid < NPOS) sN[tid] = Nv[tid];
  for (int i = tid; i < T_STEPS; i += 256) sTs[i] = ts[i];
  {
    int traj = tid >> 4, s = tid & 15;
    float v = u0s[(size_t)(trajBase + traj) * NS + s];
    sUU[0][traj * 18 + s] = v;
    out[(size_t)(trajBase + traj) * T_STEPS * NS + s] = v;
  }
  __syncthreads();  // sTs ready for forcing-table precompute

  // ---- precompute ALL forcing vectors (t-only dependence) in parallel ----
  // task = 2*step + which, 510 tasks across 256 threads (~2 each, done once,
  // removing a ~64-deep serial tanh chain from every timestep's critical path)
  for (int task = tid; task < (T_STEPS - 1) * 2; task += 256) {
    int n = task >> 1, which = task & 1;
    float t0 = sTs[n];
    float dt = sTs[n + 1] - t0;
    float tt = t0 + 0.5f * dt * (float)which;
    float f[NPOS];
    #pragma unroll
    for (int j = 0; j < NPOS; ++j) f[j] = bf2[j];
    for (int i = 0; i < HF; ++i) {
      float th = fast_tanh(tt * Wf1[i] + bf1[i]);
      #pragma unroll
      for (int j = 0; j < NPOS; ++j) f[j] += th * Wf2[i * NPOS + j];
    }
    #pragma unroll
    for (int j = 0; j < NPOS; ++j) sFall[task * NPOS + j] = f[j];
  }
  // (first barrier inside u_dot orders sFall writes before any read)

  // ---- one dH/du evaluation for all 16 trajectories; result in sDH ----
  auto u_dot = [&](int src) {
    // layer-1 forward: H1pre = U @ W1 + b1  (K = 16)
    v8f acc;
    {
      float bb = sb1[colW];
      #pragma unroll
      for (int v = 0; v < 8; ++v) acc[v] = bb;
    }
    #pragma unroll
    for (int kc = 0; kc < 4; ++kc) {
      int k0 = 4 * kc + h2x;
      v2f a, b;
      a.x = sUU[src][m16 * 18 + k0];
      a.y = sUU[src][m16 * 18 + k0 + 1];
      b.x = sW1[k0 * 132 + colW];
      b.y = sW1[(k0 + 1) * 132 + colW];
      acc = wmma_f32_k4(a, b, acc);
    }
    v8f h1v;
    #pragma unroll
    for (int v = 0; v < 8; ++v) h1v[v] = fast_tanh(acc[v]);
    #pragma unroll
    for (int v = 0; v < 8; ++v) sH1[(v + row8) * 132 + colW] = h1v[v];
    __syncthreads();

    // layer-2 forward: H2pre = H1 @ W2 + b2  (K = 128, W2 streamed from L2)
    {
      float bb = sb2[colW];
      #pragma unroll
      for (int v = 0; v < 8; ++v) acc[v] = bb;
    }
    #pragma unroll
    for (int kc = 0; kc < 32; ++kc) {
      int k0 = 4 * kc + h2x;
      v2f a, b;
      a.x = sH1[m16 * 132 + k0];
      a.y = sH1[m16 * 132 + k0 + 1];
      b.x = W2[k0 * HID + colW];
      b.y = W2[(k0 + 1) * HID + colW];
      acc = wmma_f32_k4(a, b, acc);
    }
    // d2 = (1 - tanh(H2pre)^2) * w3
    {
      float w3v = sw3[colW];
      #pragma unroll
      for (int v = 0; v < 8; ++v) {
        float th = fast_tanh(acc[v]);
        acc[v] = (1.0f - th * th) * w3v;
      }
    }
    #pragma unroll
    for (int v = 0; v < 8; ++v) sD2[(v + row8) * 132 + colW] = acc[v];
    __syncthreads();

    // layer-2 backward: G1 = d2 @ W2^T  (contiguous b64 B loads from row-major W2)
    v8f g;
    #pragma unroll
    for (int v = 0; v < 8; ++v) g[v] = 0.0f;
    #pragma unroll
    for (int kc = 0; kc < 32; ++kc) {
      int k0 = 4 * kc + h2x;
      v2f a, b;
      a.x = sD2[m16 * 132 + k0];
      a.y = sD2[m16 * 132 + k0 + 1];
      const float* wp = &W2[colW * HID + k0];
      b.x = wp[0];
      b.y = wp[1];
      g = wmma_f32_k4(a, b, g);
    }
    // d1 = (1 - h1^2) * G1   (h1 tile still in registers: same wave owns same tile)
    #pragma unroll
    for (int v = 0; v < 8; ++v) g[v] = (1.0f - h1v[v] * h1v[v]) * g[v];
    #pragma unroll
    for (int v = 0; v < 8; ++v) sH1[(v + row8) * 132 + colW] = g[v];  // overwrite h1 with d1
    __syncthreads();

    // layer-1 backward (split over waves): dH_partial = d1[k-slice] @ W1^T
    v8f p;
    #pragma unroll
    for (int v = 0; v < 8; ++v) p[v] = 0.0f;
    #pragma unroll
    for (int q = 0; q < 4; ++q) {
      int k0 = 4 * (4 * wave + q) + h2x;
      v2f a, b;
      a.x = sH1[m16 * 132 + k0];
      a.y = sH1[m16 * 132 + k0 + 1];
      b.x = sW1[m16 * 132 + k0];      // B[k][n] = W1[n][k] -> contiguous pair in row n
      b.y = sW1[m16 * 132 + k0 + 1];
      p = wmma_f32_k4(a, b, p);
    }
    #pragma unroll
    for (int v = 0; v < 8; ++v) sP[wave][(v + row8) * 17 + m16] = p[v];
    __syncthreads();

    // reduce 8 partial tiles -> sDH
    {
      int mm = tid >> 4, nn = tid & 15;
      float s = 0.0f;
      #pragma unroll
      for (int w = 0; w < 8; ++w) s += sP[w][mm * 17 + nn];
      sDH[mm * 17 + nn] = s;
    }
    __syncthreads();
  };

  // udot assembly: qdot = dH_p ; pdot = -dH_q + N.qdot + F(t)
  auto assemble = [&](int fidx) -> float {
    int traj = tid >> 4, s = tid & 15;
    if (s < NPOS) {
      return sDH[traj * 17 + NPOS + s];
    } else {
      float damp = 0.0f;
      #pragma unroll
      for (int j = 0; j < NPOS; ++j) damp += sN[j] * sDH[traj * 17 + NPOS + j];
      return -sDH[traj * 17 + (s - NPOS)] + damp + sFall[fidx * NPOS + (s - NPOS)];
    }
  };

  // ---- RK2 midpoint time loop ----
  for (int n = 0; n < T_STEPS - 1; ++n) {
    float dt = sTs[n + 1] - sTs[n];

    u_dot(0);
    {
      int traj = tid >> 4, s = tid & 15;
      float k1 = assemble(2 * n + 0);
      sUU[1][traj * 18 + s] = sUU[0][traj * 18 + s] + 0.5f * dt * k1;
    }
    __syncthreads();

    u_dot(1);
    {
      int traj = tid >> 4, s = tid & 15;
      float k2 = assemble(2 * n + 1);
      float un = sUU[0][traj * 18 + s] + dt * k2;
      out[(size_t)(trajBase + traj) * T_STEPS * NS + (size_t)(n + 1) * NS + s] = un;
      sUU[0][traj * 18 + s] = un;
    }
    __syncthreads();
  }
}

extern "C" void kernel_launch(void* const* d_in, const int* in_sizes, int n_in,
                              void* d_out, int out_size, void* d_ws, size_t ws_size,
                              hipStream_t stream) {
  (void)in_sizes; (void)n_in; (void)out_size; (void)d_ws; (void)ws_size;
  // inputs: 0 u0s, 1 t_sample, 2 W1, 3 b1, 4 W2, 5 b2, 6 W3, 7 b3 (unused: grad-invariant),
  //         8 Wf1, 9 bf1, 10 Wf2, 11 bf2, 12 N, 13 S (hardcoded symplectic structure)
  phnn_kernel<<<dim3(NTRAJ / 16), dim3(256), 0, stream>>>(
      (const float*)d_in[0], (const float*)d_in[1], (const float*)d_in[2],
      (const float*)d_in[3], (const float*)d_in[4], (const float*)d_in[5],
      (const float*)d_in[6], (const float*)d_in[8], (const float*)d_in[9],
      (const float*)d_in[10], (const float*)d_in[11], (const float*)d_in[12],
      (float*)d_out);
}